// GeneralizedDeltaRtrlLayer_56659208569008
// MI455X (gfx1250) — compile-verified
//
#include <hip/hip_runtime.h>
#include <math.h>

// ---------------------------------------------------------------------------
// GeneralizedDeltaRtrlLayer for MI455X (gfx1250, wave32)
// Shapes: IN=512, H=8, D=128, OUT=1024
// d_in order: x, S, sm_w, sm_z, sm_b, sm_v, sm_k, W_w, W_z, W_b, W_v, W_k,
//             W_q, W_o, b_o      (all float32)
// d_out: [ y(512) | S_new(131072) | sm_w_n .. sm_k_n (5 x 16777216) ]
// ---------------------------------------------------------------------------

typedef float v2f __attribute__((ext_vector_type(2)));
typedef float v8f __attribute__((ext_vector_type(8)));
typedef float f4  __attribute__((ext_vector_type(4)));

#define HH      8
#define DD      128
#define IN_SZ   512
#define OUTW    1024           // H*D
#define NROWS   5248           // 5*1024 + 128 projection rows

// workspace layout (float offsets)
#define WS_PROJ 0              // 5248: [w|z|b|v|k|q] raw projections
#define WS_AL   5248           // 1024: sigmoid(z)
#define WS_BA   6272           // 1024: be*al
#define WS_ALP  7296           // 1024: al*(1-al)
#define WS_BSD  8320           // 1024: be*s_dot
#define WS_U    9344           // 1024: delta write u
#define WS_CW   10368          // 1024: be*g*(1-g)*v
#define WS_CB   11392          // 1024: be*(1-be)*(g*v - r)
#define WS_CV   12416          // 1024: be*g
#define WS_Y0   13440          // 1024: S_new @ q (per head row)

#define OUT_Y    0
#define OUT_SNEW 512
#define OUT_SM   131584        // 512 + 8*128*128
#define SM_STRIDE 16777216     // 8*128*128*128

static __device__ __forceinline__ float sigmoidf_(float x) {
  return 1.0f / (1.0f + expf(-x));
}

// ---------------------------------------------------------------------------
// K1: projections  proj[0..5247] = concat(W_w,W_z,W_b,W_v,W_k,W_q) @ x
// One wave per 16-row tile. WMMA f32 16x16x4, K looped over 512.
// A layout (ISA 7.12.2, 32-bit A 16x4): lane l holds (M=l%16, K=2*(l/16)+g)
// in VGPR g. B is filled with x replicated across all 16 columns, so every
// column of C equals W_tile @ x; we read column N=0 from lanes 0 and 16.
// ---------------------------------------------------------------------------
__global__ __launch_bounds__(32) void k_proj(
    const float* __restrict__ x,
    const float* __restrict__ Ww, const float* __restrict__ Wz,
    const float* __restrict__ Wb, const float* __restrict__ Wv,
    const float* __restrict__ Wk, const float* __restrict__ Wq,
    float* __restrict__ ws)
{
  const int rowbase = blockIdx.x * 16;          // 0..5232
  const float* W; int lrow;
  if      (rowbase < 1024) { W = Ww; lrow = rowbase;        }
  else if (rowbase < 2048) { W = Wz; lrow = rowbase - 1024; }
  else if (rowbase < 3072) { W = Wb; lrow = rowbase - 2048; }
  else if (rowbase < 4096) { W = Wv; lrow = rowbase - 3072; }
  else if (rowbase < 5120) { W = Wk; lrow = rowbase - 4096; }
  else                     { W = Wq; lrow = rowbase - 5120; }

  const int l     = threadIdx.x;                // 0..31
  const int m     = l & 15;                     // A row in tile
  const int khalf = (l >> 4) * 2;               // K offset of this half-wave

#if __has_builtin(__builtin_amdgcn_wmma_f32_16x16x4_f32)
  const float* wrow = W + (size_t)(lrow + m) * IN_SZ;
  v8f c = {0.f, 0.f, 0.f, 0.f, 0.f, 0.f, 0.f, 0.f};
  for (int kk = 0; kk < IN_SZ; kk += 4) {
    v2f a, b;
    a.x = wrow[kk + khalf];
    a.y = wrow[kk + khalf + 1];
    b.x = x[kk + khalf];                        // replicate x across all N
    b.y = x[kk + khalf + 1];
    c = __builtin_amdgcn_wmma_f32_16x16x4_f32(
        false, a, false, b, (short)0, c, false, false);
  }
  if (m == 0) {                                 // lanes 0 (M 0..7), 16 (M 8..15)
    const int mbase = (l >> 4) * 8;
    #pragma unroll
    for (int r = 0; r < 8; ++r)
      ws[WS_PROJ + rowbase + mbase + r] = c[r];
  }
#else
  // scalar fallback (should not trigger on gfx1250)
  if (l < 16) {
    const float* wr = W + (size_t)(lrow + l) * IN_SZ;
    float acc = 0.f;
    for (int kk = 0; kk < IN_SZ; ++kk) acc += wr[kk] * x[kk];
    ws[WS_PROJ + rowbase + l] = acc;
  }
  (void)khalf; (void)m;
#endif
}

// ---------------------------------------------------------------------------
// K2: gates, s_dot, u, S_new (to d_out), y0, per-row coefficients (to ws).
// One wave per (h,i) row; 1024 rows -> 128 blocks x 256 threads.
// ---------------------------------------------------------------------------
__global__ __launch_bounds__(256) void k_state(
    const float* __restrict__ S, float* __restrict__ out,
    float* __restrict__ ws)
{
  const int row  = blockIdx.x * 8 + (threadIdx.x >> 5);  // 0..1023 = h*128+i
  const int lane = threadIdx.x & 31;
  const int h    = row >> 7;

  const float wr = ws[WS_PROJ + row];
  const float zr = ws[WS_PROJ + 1024 + row];
  const float br = ws[WS_PROJ + 2048 + row];
  const float vr = ws[WS_PROJ + 3072 + row];
  const float g  = sigmoidf_(wr);
  const float al = sigmoidf_(zr);
  const float be = sigmoidf_(br);

  const f4* Srow = (const f4*)(S + (size_t)row * DD);
  const f4* krow = (const f4*)(ws + WS_PROJ + 4096 + h * DD);
  const f4* qrow = (const f4*)(ws + WS_PROJ + 5120);

  f4 s4 = Srow[lane];
  f4 k4 = krow[lane];
  float sd = s4.x * k4.x + s4.y * k4.y + s4.z * k4.z + s4.w * k4.w;
  #pragma unroll
  for (int off = 16; off; off >>= 1) sd += __shfl_xor(sd, off, 32);

  const float r = al * sd;
  const float u = be * (g * vr - r);

  f4 sn;
  sn.x = al * s4.x + u * k4.x;
  sn.y = al * s4.y + u * k4.y;
  sn.z = al * s4.z + u * k4.z;
  sn.w = al * s4.w + u * k4.w;
  ((f4*)(out + OUT_SNEW + (size_t)row * DD))[lane] = sn;

  f4 q4 = qrow[lane];
  float yp = sn.x * q4.x + sn.y * q4.y + sn.z * q4.z + sn.w * q4.w;
  #pragma unroll
  for (int off = 16; off; off >>= 1) yp += __shfl_xor(yp, off, 32);

  if (lane == 0) {
    ws[WS_AL  + row] = al;
    ws[WS_BA  + row] = be * al;
    ws[WS_ALP + row] = al * (1.f - al);
    ws[WS_BSD + row] = be * sd;
    ws[WS_U   + row] = u;
    ws[WS_CW  + row] = be * (g * (1.f - g)) * vr;
    ws[WS_CB  + row] = (be * (1.f - be)) * (g * vr - r);
    ws[WS_CV  + row] = be * g;
    ws[WS_Y0  + row] = yp;
  }
}

// ---------------------------------------------------------------------------
// K3: y = W_o @ y0 + b_o.  One wave per output (512) -> 64 blocks.
// ---------------------------------------------------------------------------
__global__ __launch_bounds__(256) void k_out(
    const float* __restrict__ Wo, const float* __restrict__ bo,
    const float* __restrict__ ws, float* __restrict__ out)
{
  const int o    = blockIdx.x * 8 + (threadIdx.x >> 5);  // 0..511
  const int lane = threadIdx.x & 31;
  const f4* wr = (const f4*)(Wo + (size_t)o * OUTW);
  const f4* y0 = (const f4*)(ws + WS_Y0);
  float acc = 0.f;
  #pragma unroll
  for (int s = 0; s < 8; ++s) {
    f4 a = wr[s * 32 + lane];
    f4 b = y0[s * 32 + lane];
    acc += a.x * b.x + a.y * b.y + a.z * b.z + a.w * b.w;
  }
  #pragma unroll
  for (int off = 16; off; off >>= 1) acc += __shfl_xor(acc, off, 32);
  if (lane == 0) out[OUT_Y + o] = acc + bo[o];
}

// ---------------------------------------------------------------------------
// K4: the bandwidth kernel. grid = (1024 slabs, 5 tensors), 256 threads.
// Per block: stream one 128x128 slab (64 KB) into registers (NT loads),
// reduce contr[c] = sum_q k[q]*P[q,c] via conflict-free LDS staging,
// second pass from registers applies rec() + direct Jacobian, NT stores.
// Thread t owns c-quad c0=(t%32)*4 (constant over s) and rows j=8s+t/32.
// ---------------------------------------------------------------------------
__global__ __launch_bounds__(256) void k_rtrl(
    const float* __restrict__ Pw, const float* __restrict__ Pz,
    const float* __restrict__ Pb, const float* __restrict__ Pv,
    const float* __restrict__ Pk,
    const float* __restrict__ S,
    float* __restrict__ out, const float* __restrict__ ws)
{
  __shared__ float k_lds[128];
  __shared__ float Srow[128];
  __shared__ float contr[128];
  __shared__ float part[8 * 128];

  const int slab = blockIdx.x;         // h*128 + i
  const int type = blockIdx.y;         // 0..4 -> w,z,b,v,k
  const int h    = slab >> 7;
  const int i    = slab & 127;
  const int t    = threadIdx.x;

  const float* Pin;
  if      (type == 0) Pin = Pw;
  else if (type == 1) Pin = Pz;
  else if (type == 2) Pin = Pb;
  else if (type == 3) Pin = Pv;
  else                Pin = Pk;
  const float* slabP = Pin + (size_t)slab * (DD * DD);
  float*       slabO = out + OUT_SM + (size_t)type * SM_STRIDE
                           + (size_t)slab * (DD * DD);
  __builtin_prefetch(slabP, 0, 1);     // global_prefetch_b8

  if (t < 128) {
    k_lds[t] = ws[WS_PROJ + 4096 + h * DD + t];
    Srow[t]  = S[(size_t)slab * DD + t];
  }
  __syncthreads();

  const int wv = t >> 5;               // wave id 0..7 -> row group
  const int c0 = (t & 31) << 2;        // c-quad base, constant over s

  const f4* src = (const f4*)slabP;
  f4 vals[16];
  float p0 = 0.f, p1 = 0.f, p2 = 0.f, p3 = 0.f;
  #pragma unroll
  for (int s = 0; s < 16; ++s) {
    f4 v = __builtin_nontemporal_load(src + (s * 256 + t));
    vals[s] = v;
    const float kq = k_lds[(s << 3) + wv];   // q = 8s + t/32
    p0 += kq * v.x; p1 += kq * v.y; p2 += kq * v.z; p3 += kq * v.w;
  }
  // conflict-free staging: within a wave, threads own disjoint c-quads
  part[wv * 128 + c0 + 0] = p0;
  part[wv * 128 + c0 + 1] = p1;
  part[wv * 128 + c0 + 2] = p2;
  part[wv * 128 + c0 + 3] = p3;
  __syncthreads();
  if (t < 128) {
    float s8 = 0.f;
    #pragma unroll
    for (int w = 0; w < 8; ++w) s8 += part[w * 128 + t];
    contr[t] = s8;
  }
  __syncthreads();

  const float al_i = ws[WS_AL + slab];
  const float ba_i = ws[WS_BA + slab];
  f4 cv; cv.x = contr[c0]; cv.y = contr[c0 + 1];
         cv.z = contr[c0 + 2]; cv.w = contr[c0 + 3];

  // per-type direct-Jacobian state
  const bool hascol = (type < 4) && (i >= c0) && (i < c0 + 4);  // column c==i
  const int  colm   = i - c0;
  float ccoef = 0.f, alp = 0.f, bsd = 0.f, uu = 0.f;
  if      (type == 0) ccoef = ws[WS_CW + slab];
  else if (type == 1) { alp = ws[WS_ALP + slab]; bsd = ws[WS_BSD + slab]; }
  else if (type == 2) ccoef = ws[WS_CB + slab];
  else if (type == 3) ccoef = ws[WS_CV + slab];
  else                uu    = ws[WS_U + slab];
  f4 sC; sC.x = Srow[c0]; sC.y = Srow[c0 + 1];
         sC.z = Srow[c0 + 2]; sC.w = Srow[c0 + 3];

  f4* dst = (f4*)slabO;
  #pragma unroll
  for (int s = 0; s < 16; ++s) {
    const int   j  = (s << 3) + wv;
    const float kj = k_lds[j];
    const float bk = ba_i * kj;
    f4 v = vals[s];
    f4 o;
    o.x = al_i * v.x - bk * cv.x;
    o.y = al_i * v.y - bk * cv.y;
    o.z = al_i * v.z - bk * cv.z;
    o.w = al_i * v.w - bk * cv.w;
    if (type == 4) {
      // D_k = u*delta_{jc} - ba_i * S[h,i,c] * k_j
      o.x -= bk * sC.x; o.y -= bk * sC.y; o.z -= bk * sC.z; o.w -= bk * sC.w;
      if (j >= c0 && j < c0 + 4) {
        const int mm = j - c0;
        if      (mm == 0) o.x += uu;
        else if (mm == 1) o.y += uu;
        else if (mm == 2) o.z += uu;
        else              o.w += uu;
      }
    } else if (hascol) {
      float add;
      if (type == 1) add = alp * (Srow[j] - bsd * kj);   // D_z column
      else           add = ccoef * kj;                   // D_w / D_b / D_v
      if      (colm == 0) o.x += add;
      else if (colm == 1) o.y += add;
      else if (colm == 2) o.z += add;
      else                o.w += add;
    }
    __builtin_nontemporal_store(o, dst + (s * 256 + t));
  }
}

// ---------------------------------------------------------------------------
extern "C" void kernel_launch(void* const* d_in, const int* in_sizes, int n_in,
                              void* d_out, int out_size, void* d_ws,
                              size_t ws_size, hipStream_t stream) {
  (void)in_sizes; (void)n_in; (void)out_size; (void)ws_size;
  const float* x   = (const float*)d_in[0];
  const float* S   = (const float*)d_in[1];
  const float* smw = (const float*)d_in[2];
  const float* smz = (const float*)d_in[3];
  const float* smb = (const float*)d_in[4];
  const float* smv = (const float*)d_in[5];
  const float* smk = (const float*)d_in[6];
  const float* Ww  = (const float*)d_in[7];
  const float* Wz  = (const float*)d_in[8];
  const float* Wb  = (const float*)d_in[9];
  const float* Wv  = (const float*)d_in[10];
  const float* Wk  = (const float*)d_in[11];
  const float* Wq  = (const float*)d_in[12];
  const float* Wo  = (const float*)d_in[13];
  const float* bo  = (const float*)d_in[14];
  float* out = (float*)d_out;
  float* ws  = (float*)d_ws;   // needs 14464 floats (~58 KB)

  k_proj <<<NROWS / 16, 32, 0, stream>>>(x, Ww, Wz, Wb, Wv, Wk, Wq, ws);
  k_state<<<128, 256, 0, stream>>>(S, out, ws);
  k_out  <<<64, 256, 0, stream>>>(Wo, bo, ws, out);
  k_rtrl <<<dim3(1024, 5), 256, 0, stream>>>(smw, smz, smb, smv, smk,
                                             S, out, ws);
}